// TopKLogitProcessor_9483287790145
// MI455X (gfx1250) — compile-verified
//
#include <hip/hip_runtime.h>

#define THREADS 256
#define BINS    4096
#define GROUPS  (BINS / THREADS)   // 16 bins per thread in the scan
#define CAP     3072               // tie-candidate list capacity
#define NBUF    4                  // async pipeline depth (3 loads in flight)

#if defined(__HIP_DEVICE_COMPILE__) && defined(__gfx1250__)
#define HAVE_ASYNC 1
#endif

typedef float v4f __attribute__((ext_vector_type(4)));

// Monotonic float -> u32 key: larger float <=> larger key (finite inputs).
__device__ __forceinline__ unsigned keyOf(float f) {
    unsigned u = __float_as_uint(f);
    return u ^ ((unsigned)((int)u >> 31) | 0x80000000u);
}

// CDNA5 async copy engine: 16B global -> LDS per lane, tracked by ASYNCcnt.
// Default (RT) temporal policy on purpose: this read should stay in L2 so the
// reverse-order second pass can hit it.
__device__ __forceinline__ void async_copy16(float* ldsDst, const float* gSrc) {
#if defined(HAVE_ASYNC)
    unsigned l = (unsigned)(unsigned long long)(__attribute__((address_space(3))) float*)ldsDst;
    unsigned long long g = (unsigned long long)gSrc;
    asm volatile("global_load_async_to_lds_b128 %0, %1, off"
                 :: "v"(l), "v"(g) : "memory");
#else
    *(v4f*)ldsDst = *(const v4f*)gSrc;
#endif
}

__device__ __forceinline__ void async_wait(int rem) {
#if defined(HAVE_ASYNC)
    if (rem >= 3)      asm volatile("s_wait_asynccnt 3" ::: "memory");
    else if (rem == 2) asm volatile("s_wait_asynccnt 2" ::: "memory");
    else if (rem == 1) asm volatile("s_wait_asynccnt 1" ::: "memory");
    else               asm volatile("s_wait_asynccnt 0" ::: "memory");
#else
    (void)rem;
#endif
}

__global__ __launch_bounds__(THREADS)
void TopKLogitProcessor_9483287790145_kernel(const float* __restrict__ logits,
                                             const int*   __restrict__ topk_ptr,
                                             float*       __restrict__ out,
                                             int V) {
    __shared__ unsigned hist[BINS];                          // 16 KB
    __shared__ __align__(16) float stage[NBUF][THREADS * 4]; // 16 KB quad buffer
    __shared__ float   listVal[CAP];                         // 12 KB
    __shared__ int     listIdx[CAP];                         // 12 KB
    __shared__ unsigned partialS[THREADS];
    __shared__ unsigned prefixS[THREADS];
    __shared__ int      sBucket;
    __shared__ unsigned sGreater;
    __shared__ unsigned sCount;

    const int tid = threadIdx.x;
    const int row = blockIdx.x;
    const float* rowp = logits + (long long)row * V;
    float*       outp = out    + (long long)row * V;
    const unsigned K = (unsigned)(*topk_ptr);

    for (int i = tid; i < BINS; i += THREADS) hist[i] = 0u;
    if (tid == 0) sCount = 0u;
    __syncthreads();

    const int TILE = THREADS * 4;          // 1024 floats per block-tile
    const int fullTiles = V / TILE;        // 125 for V=128000 (exact)

    // ---------- Pass 1: 12-bit histogram, async 4-deep pipelined streaming ----------
    {
        const int pre = (fullTiles < NBUF - 1) ? fullTiles : (NBUF - 1);
        for (int p = 0; p < pre; ++p)
            async_copy16(&stage[p & (NBUF - 1)][tid * 4], rowp + p * TILE + tid * 4);

        for (int t = 0; t < fullTiles; ++t) {
            const int nx = t + (NBUF - 1);
            if (nx < fullTiles)
                async_copy16(&stage[nx & (NBUF - 1)][tid * 4], rowp + nx * TILE + tid * 4);
            async_wait(fullTiles - 1 - t);        // tile t guaranteed landed
            v4f v = *(const v4f*)&stage[t & (NBUF - 1)][tid * 4];
            atomicAdd(&hist[keyOf(v.x) >> 20], 1u);
            atomicAdd(&hist[keyOf(v.y) >> 20], 1u);
            atomicAdd(&hist[keyOf(v.z) >> 20], 1u);
            atomicAdd(&hist[keyOf(v.w) >> 20], 1u);
        }
    }
    for (int i = fullTiles * TILE + tid; i < V; i += THREADS)   // tail (none for 128000)
        atomicAdd(&hist[keyOf(rowp[i]) >> 20], 1u);
    __syncthreads();

    // ---------- Find threshold bucket: descending suffix scan over 4096 bins ----------
    unsigned psum = 0;
    const int base = BINS - 1 - tid * GROUPS;   // this thread's highest bin
    #pragma unroll
    for (int i = 0; i < GROUPS; ++i) psum += hist[base - i];
    partialS[tid] = psum;
    __syncthreads();
    if (tid == 0) {
        unsigned acc = 0;
        for (int i = 0; i < THREADS; ++i) { prefixS[i] = acc; acc += partialS[i]; }
    }
    __syncthreads();
    const unsigned pre2 = prefixS[tid];
    if (pre2 < K && pre2 + psum >= K) {         // unique owner of the crossing
        unsigned cum = pre2;
        for (int i = 0; i < GROUPS; ++i) {
            const int b = base - i;
            const unsigned c = hist[b];
            if (cum + c >= K) { sBucket = b; sGreater = cum; break; }
            cum += c;
        }
    }
    __syncthreads();
    const int      Bk      = sBucket;
    const unsigned greater = sGreater;

    // ---------- Pass 2: fused masked write + tie collection ----------
    // Reverse tile order: the row's tail was read most recently in pass 1, so it
    // is the part most likely still resident in the 192MB L2. Loads are marked
    // non-temporal (last use); stores are non-temporal so the 512MB output
    // stream does not evict logits from L2.
    const float NEG_INF = __int_as_float((int)0xFF800000);
    for (int i = fullTiles * TILE + tid; i < V; i += THREADS) {  // tail first
        const float f   = __builtin_nontemporal_load(rowp + i);
        const int   bin = (int)(keyOf(f) >> 20);
        __builtin_nontemporal_store((bin > Bk) ? f : NEG_INF, outp + i);
        if (bin == Bk) {
            unsigned p = atomicAdd(&sCount, 1u);
            if (p < CAP) { listVal[p] = f; listIdx[p] = i; }
        }
    }
    for (int t = fullTiles - 1; t >= 0; --t) {
        const int idx = t * TILE + tid * 4;
        v4f v = __builtin_nontemporal_load((const v4f*)(rowp + idx));
        v4f o;
        #pragma unroll
        for (int c = 0; c < 4; ++c) {
            const float f   = v[c];
            const int   bin = (int)(keyOf(f) >> 20);
            o[c] = (bin > Bk) ? f : NEG_INF;
            if (bin == Bk) {
                unsigned p = atomicAdd(&sCount, 1u);
                if (p < CAP) { listVal[p] = f; listIdx[p] = idx + c; }
            }
        }
        __builtin_nontemporal_store(o, (v4f*)(outp + idx));
    }
    __syncthreads();

    // ---------- Rank-select k' = K - greater ties; overwrite their -inf ----------
    const int m  = (int)(sCount < (unsigned)CAP ? sCount : (unsigned)CAP);
    const int kp = (int)K - (int)greater;
    for (int e = tid; e < m; e += THREADS) {
        const float ve = listVal[e];
        const int   ie = listIdx[e];
        int rank = 0;
        for (int j = 0; j < m; ++j) {
            const float vj = listVal[j];
            rank += (vj > ve) || ((vj == ve) && (listIdx[j] < ie));
        }
        if (rank < kp) outp[ie] = ve;   // stable tie-break matches lax.top_k
    }
}

extern "C" void kernel_launch(void* const* d_in, const int* in_sizes, int n_in,
                              void* d_out, int out_size, void* d_ws, size_t ws_size,
                              hipStream_t stream) {
    const float* logits = (const float*)d_in[0];
    // d_in[1] = position_ids (unused, matching the reference)
    const int* topk = (const int*)d_in[2];
    float* out = (float*)d_out;

    const int B = in_sizes[1];               // 1024 rows
    const int V = in_sizes[0] / B;           // 128000 vocab

    dim3 grid(B), block(THREADS);
    TopKLogitProcessor_9483287790145_kernel<<<grid, block, 0, stream>>>(logits, topk, out, V);
}